// GATConv_54623394070822
// MI455X (gfx1250) — compile-verified
//
#include <hip/hip_runtime.h>
#include <hip/hip_bf16.h>

#define EMB 64
#define HEADS 2
#define NEG_SLOPE 0.2f

typedef __attribute__((ext_vector_type(2))) float v2f;
typedef __attribute__((ext_vector_type(8))) float v8f;

// ---------------------------------------------------------------------------
// float atomic max via monotonic integer-bit trick (correct for mixed signs
// when initialized to -inf).
__device__ __forceinline__ void atomicMaxF(float* addr, float val) {
    if (val >= 0.0f) {
        atomicMax((int*)addr, __float_as_int(val));
    } else {
        atomicMin((unsigned int*)addr, __float_as_uint(val));
    }
}

// ---------------------------------------------------------------------------
// K0: initialize output with bias, amax with -inf, denom with 0.
__global__ void k_init(float* __restrict__ out, const float* __restrict__ bias,
                       float* __restrict__ amax, float* __restrict__ denom, int N) {
    const int tid = blockIdx.x * blockDim.x + threadIdx.x;
    if (tid < N * EMB) out[tid] = bias[tid & (EMB - 1)];
    if (tid < N * HEADS) {
        amax[tid]  = __int_as_float(0xff800000);  // -inf
        denom[tid] = 0.0f;
    }
}

// ---------------------------------------------------------------------------
// K1: h = x @ W_lin + b_lin via V_WMMA_F32_16X16X4_F32 (full fp32 precision).
// One wave computes one 16x16 tile of the N x 128 output; K loop = 64/4 = 16
// WMMA issues. A-operand layout (16x4 f32, 2 VGPRs): lanes 0-15 hold K=k0,k0+1,
// lanes 16-31 hold K=k0+2,k0+3 for rows M=lane&15. B mirrors with rows=K.
__global__ __launch_bounds__(32) void k_gemm_lin(const float* __restrict__ X,
                                                 const float* __restrict__ W,
                                                 const float* __restrict__ B,
                                                 float* __restrict__ H, int N) {
    const int rowTile = blockIdx.x >> 3;   // 128/16 = 8 column tiles
    const int colTile = blockIdx.x & 7;
    const int lane = threadIdx.x;
    const int l15  = lane & 15;
    const int half = lane >> 4;            // 0 or 1
    const int row0 = rowTile * 16;
    const int col0 = colTile * 16;

    const int rA = min(row0 + l15, N - 1); // clamp (WMMA needs full EXEC)
    const float* xrow = X + (size_t)rA * EMB;
    const int kA = half * 2;

    v8f acc = {};
#pragma unroll
    for (int k0 = 0; k0 < EMB; k0 += 4) {
        v2f a, b;
        a.x = xrow[k0 + kA];
        a.y = xrow[k0 + kA + 1];
        b.x = W[(size_t)(k0 + kA)     * 128 + col0 + l15];
        b.y = W[(size_t)(k0 + kA + 1) * 128 + col0 + l15];
        acc = __builtin_amdgcn_wmma_f32_16x16x4_f32(
            /*neg_a=*/false, a, /*neg_b=*/false, b,
            /*c_mod=*/(short)0, acc, /*reuse_a=*/false, /*reuse_b=*/false);
    }

    const float bv = B[col0 + l15];
#pragma unroll
    for (int r = 0; r < 8; ++r) {
        const int row = row0 + r + half * 8;   // C/D layout: lanes16-31 -> M+8
        if (row < N) H[(size_t)row * 128 + col0 + l15] = acc[r] + bv;
    }
}

// ---------------------------------------------------------------------------
// K2: per-node attention scores s_i[n,h] = h[n,h,:]·att_i[h],
//     s_j[n,h] = h[n,h,:]·att_j[h].  One wave per node.
__global__ __launch_bounds__(128) void k_node_scores(const float* __restrict__ H,
                                                     const float* __restrict__ att,
                                                     float* __restrict__ s_i,
                                                     float* __restrict__ s_j, int N) {
    const int lane = threadIdx.x & 31;
    const int n = blockIdx.x * 4 + (threadIdx.x >> 5);
    if (n >= N) return;
    float si0 = 0.f, si1 = 0.f, sj0 = 0.f, sj1 = 0.f;
#pragma unroll
    for (int t = 0; t < 4; ++t) {
        const int idx = lane + t * 32;       // 0..127 over the node row
        const float v = H[(size_t)n * 128 + idx];
        const int head = idx >> 6;
        const int c = idx & 63;
        const float ai = att[head * 128 + c];
        const float aj = att[head * 128 + 64 + c];
        if (head == 0) { si0 += v * ai; sj0 += v * aj; }
        else           { si1 += v * ai; sj1 += v * aj; }
    }
#pragma unroll
    for (int off = 16; off > 0; off >>= 1) {
        si0 += __shfl_xor(si0, off, 32);
        si1 += __shfl_xor(si1, off, 32);
        sj0 += __shfl_xor(sj0, off, 32);
        sj1 += __shfl_xor(sj1, off, 32);
    }
    if (lane == 0) {
        s_i[n * 2 + 0] = si0; s_i[n * 2 + 1] = si1;
        s_j[n * 2 + 0] = sj0; s_j[n * 2 + 1] = sj1;
    }
}

// ---------------------------------------------------------------------------
// K3: fold W_edge/b_edge against att_j:  wEA[h][k] = sum_c W_edge[k, h*64+c]*att_j[h][c]
__global__ void k_wea(const float* __restrict__ We, const float* __restrict__ be,
                      const float* __restrict__ att,
                      float* __restrict__ wEA, float* __restrict__ bEA) {
    const int t = threadIdx.x;
    if (t < 10) {
        const int h = t / 5, k = t % 5;
        float s = 0.f;
        for (int c = 0; c < EMB; ++c)
            s += We[k * 128 + h * 64 + c] * att[h * 128 + 64 + c];
        wEA[h * 5 + k] = s;
    } else if (t < 12) {
        const int h = t - 10;
        float s = 0.f;
        for (int c = 0; c < EMB; ++c)
            s += be[h * 64 + c] * att[h * 128 + 64 + c];
        bEA[h] = s;
    }
}

// ---------------------------------------------------------------------------
// K4: per-edge logits + leaky-ReLU + atomic segment max over src.
__global__ void k_alpha(const long long* __restrict__ ei, const float* __restrict__ EA,
                        const float* __restrict__ s_i, const float* __restrict__ s_j,
                        const float* __restrict__ wEA, const float* __restrict__ bEA,
                        float* __restrict__ alpha, float* __restrict__ amax, int E) {
    const int e = blockIdx.x * blockDim.x + threadIdx.x;
    if (e >= E) return;
    const long long src = ei[e];
    const long long dst = ei[(size_t)E + e];
    float a[5];
#pragma unroll
    for (int k = 0; k < 5; ++k) a[k] = EA[(size_t)e * 5 + k];
#pragma unroll
    for (int h = 0; h < HEADS; ++h) {
        float al = s_i[src * 2 + h] + s_j[dst * 2 + h] + bEA[h];
#pragma unroll
        for (int k = 0; k < 5; ++k) al += a[k] * wEA[h * 5 + k];
        al = (al >= 0.f) ? al : NEG_SLOPE * al;
        alpha[(size_t)e * 2 + h] = al;
        atomicMaxF(&amax[src * 2 + h], al);
    }
}

// ---------------------------------------------------------------------------
// K5: e = exp(alpha - amax[src]); denom[src] += e.
__global__ void k_expsum(const long long* __restrict__ ei, const float* __restrict__ alpha,
                         const float* __restrict__ amax, float* __restrict__ eexp,
                         float* __restrict__ denom, int E) {
    const int e = blockIdx.x * blockDim.x + threadIdx.x;
    if (e >= E) return;
    const long long src = ei[e];
#pragma unroll
    for (int h = 0; h < HEADS; ++h) {
        const float v = expf(alpha[(size_t)e * 2 + h] - amax[src * 2 + h]);
        eexp[(size_t)e * 2 + h] = v;
        atomicAdd(&denom[src * 2 + h], v);
    }
}

// ---------------------------------------------------------------------------
// K6: fused message pass. ee is recomputed from W_edge (held in LDS) instead
// of materializing the 410 MB E x 128 intermediate. One thread per (edge,c);
// heads fused and averaged, then a single scattered atomicAdd per output elem.
__global__ __launch_bounds__(256) void k_message(const long long* __restrict__ ei,
                                                 const float* __restrict__ EA,
                                                 const float* __restrict__ H,
                                                 const float* __restrict__ We,
                                                 const float* __restrict__ be,
                                                 const float* __restrict__ eexp,
                                                 const float* __restrict__ denom,
                                                 float* __restrict__ out, int E) {
    __shared__ float sW[5 * 128 + 128];
    for (int i = threadIdx.x; i < 768; i += blockDim.x)
        sW[i] = (i < 640) ? We[i] : be[i - 640];
    __syncthreads();

    const int tid = blockIdx.x * blockDim.x + threadIdx.x;
    const int e = tid >> 6;
    const int c = tid & 63;
    if (e >= E) return;
    const long long src = ei[e];
    const long long dst = ei[(size_t)E + e];
    float a[5];
#pragma unroll
    for (int k = 0; k < 5; ++k) a[k] = EA[(size_t)e * 5 + k];

    float m = 0.f;
#pragma unroll
    for (int h = 0; h < HEADS; ++h) {
        const float w = eexp[(size_t)e * 2 + h] / (denom[src * 2 + h] + 1e-16f);
        float ee = sW[640 + h * 64 + c];
#pragma unroll
        for (int k = 0; k < 5; ++k) ee += a[k] * sW[k * 128 + h * 64 + c];
        const float xj = H[(size_t)dst * 128 + h * 64 + c] + ee;
        m += xj * w;
    }
    atomicAdd(&out[src * 64 + c], 0.5f * m);   // mean over 2 heads
}

// ---------------------------------------------------------------------------
extern "C" void kernel_launch(void* const* d_in, const int* in_sizes, int n_in,
                              void* d_out, int out_size, void* d_ws, size_t ws_size,
                              hipStream_t stream) {
    const float*     x      = (const float*)d_in[0];
    const long long* ei     = (const long long*)d_in[1];   // int64 edge_index (2,E)
    const float*     ea     = (const float*)d_in[2];
    const float*     W_lin  = (const float*)d_in[3];
    const float*     b_lin  = (const float*)d_in[4];
    const float*     W_edge = (const float*)d_in[5];
    const float*     b_edge = (const float*)d_in[6];
    const float*     att    = (const float*)d_in[7];
    const float*     bias   = (const float*)d_in[8];
    float* out = (float*)d_out;

    const int N = in_sizes[0] / EMB;
    const int E = in_sizes[1] / 2;

    // Workspace layout (floats): h[N*128] | s_i[N*2] | s_j[N*2] | amax[N*2]
    //                          | denom[N*2] | wEA[16] | bEA[16] | alpha[E*2] | eexp[E*2]
    float* ws    = (float*)d_ws;
    float* hbuf  = ws;
    float* s_i   = hbuf  + (size_t)N * 128;
    float* s_j   = s_i   + (size_t)N * 2;
    float* amax  = s_j   + (size_t)N * 2;
    float* denom = amax  + (size_t)N * 2;
    float* wEA   = denom + (size_t)N * 2;
    float* bEA   = wEA + 16;
    float* alpha = bEA + 16;
    float* eexp  = alpha + (size_t)E * 2;

    k_init<<<(N * EMB + 255) / 256, 256, 0, stream>>>(out, bias, amax, denom, N);

    const int rowTiles = (N + 15) / 16;
    k_gemm_lin<<<rowTiles * 8, 32, 0, stream>>>(x, W_lin, b_lin, hbuf, N);

    k_node_scores<<<(N + 3) / 4, 128, 0, stream>>>(hbuf, att, s_i, s_j, N);
    k_wea<<<1, 32, 0, stream>>>(W_edge, b_edge, att, wEA, bEA);
    k_alpha<<<(E + 255) / 256, 256, 0, stream>>>(ei, ea, s_i, s_j, wEA, bEA, alpha, amax, E);
    k_expsum<<<(E + 255) / 256, 256, 0, stream>>>(ei, alpha, amax, eexp, denom, E);

    const long long msgThreads = (long long)E * 64;
    k_message<<<(int)((msgThreads + 255) / 256), 256, 0, stream>>>(
        ei, ea, hbuf, W_edge, b_edge, eexp, denom, out, E);
}